// FTCLLoss_51711406244502
// MI455X (gfx1250) — compile-verified
//
#include <hip/hip_runtime.h>
#include <math.h>

// ---------------- problem constants ----------------
#define NB    64        // batch
#define NC    20        // NCLS
#define NC1   21        // NCLS+1
#define TT    2048      // T
#define DD    2048      // D
#define LCSL  64
#define FSDL  32
#define LAMB1 1.0f
#define LAMB2 5e-05f
#define LAMB3 8e-05f
#define LAMB_LCS 0.01f
#define LAMB_FSD 0.01f
#define GAMMA 0.1f
#define THRES 0.5f

// ---------------- ws layout (floats) ----------------
#define W_ACC    0       // 1    running scalar loss accumulator
#define W_PL     64      // 64   pair_label
#define W_LCSV   128     // 64   per-batch lcs dp value
#define W_FSDV   192     // 64   per-batch fsd dp value
#define W_LCSN0  256     // 4096 row norms lcs_candi_0
#define W_LCSN1  4352    // 4096 row norms lcs_candi_1
#define W_FSDN   8448    // 3*4096 half-row norms: [act0][act1][bak1], idx=(row*2+half)

// ---------------- WMMA types ----------------
typedef __attribute__((ext_vector_type(2))) float v2f;
typedef __attribute__((ext_vector_type(8))) float v8f;

__device__ __forceinline__ v8f wmma4(v2f a, v2f b, v8f c) {
#if defined(__gfx1250__)
  // D(16x16,f32) += A(16x4,f32) * B(4x16,f32)
  return __builtin_amdgcn_wmma_f32_16x16x4_f32(false, a, false, b, (short)0, c,
                                               false, false);
#else
  // host-pass placeholder only
  c[0] += a[0] * b[0] + a[1] * b[1];
  return c;
#endif
}

// ---------------- reductions ----------------
__device__ __forceinline__ float wave_sum(float v) {
  for (int o = 16; o > 0; o >>= 1) v += __shfl_down(v, o, 32);
  return v;
}

__device__ __forceinline__ float bred(float v, float* sh, int n) {
  int t = threadIdx.x;
  sh[t] = v;
  __syncthreads();
  for (int k = n >> 1; k > 0; k >>= 1) {
    if (t < k) sh[t] += sh[t + k];
    __syncthreads();
  }
  float r = sh[0];
  __syncthreads();
  return r;
}

// ---------------- kernels ----------------
__global__ void k_init(float* ws) {
  if (threadIdx.x == 0 && blockIdx.x == 0) ws[W_ACC] = 0.f;
}

__global__ void k_pair(const float* __restrict__ vl0,
                       const float* __restrict__ vl1, float* ws) {
  int b = threadIdx.x;
  if (b < NB) {
    float s = 0.f;
    for (int c = 0; c < NC; ++c) s += vl0[b * NC + c] * vl1[b * NC + c];
    ws[W_PL + b] = (s > 0.f) ? 1.f : 0.f;
  }
}

// One wave per row-norm job. jobs: 4096 lcs0 rows, 4096 lcs1 rows,
// then 3*4096 fsd half rows (len 1024).
__global__ __launch_bounds__(256) void k_norms(const float* __restrict__ lcs0,
                                               const float* __restrict__ lcs1,
                                               const float* __restrict__ fa0,
                                               const float* __restrict__ fa1,
                                               const float* __restrict__ fb1,
                                               float* ws) {
  int wave = threadIdx.x >> 5, lane = threadIdx.x & 31;
  long job = (long)blockIdx.x * 8 + wave;
  const float* p;
  float* out;
  int len;
  if (job < 4096) {
    p = lcs0 + job * (long)DD; out = ws + W_LCSN0 + job; len = DD;
  } else if (job < 8192) {
    long r = job - 4096;
    p = lcs1 + r * (long)DD; out = ws + W_LCSN1 + r; len = DD;
  } else {
    long fj = job - 8192;
    int arr = (int)(fj / 4096);
    long rem = fj - (long)arr * 4096;
    long row = rem >> 1;
    int half = (int)(rem & 1);
    const float* base = (arr == 0) ? fa0 : ((arr == 1) ? fa1 : fb1);
    p = base + row * (long)DD + half * 1024;
    out = ws + W_FSDN + arr * 4096 + rem;
    len = 1024;
  }
  float s = 0.f;
  const float4* p4 = (const float4*)p;
  int nv = len >> 2;
  for (int i = lane; i < nv; i += 32) {
    float4 v = p4[i];
    s += v.x * v.x + v.y * v.y + v.z * v.z + v.w * v.w;
  }
  s = wave_sum(s);
  if (lane == 0) *out = sqrtf(s);
}

__global__ void k_cls(const float* pi0, const float* pc0, const float* pb0,
                      const float* vl0, const float* pi1, const float* pc1,
                      const float* pb1, const float* vl1, float* ws) {
  __shared__ float red[64];
  int b = threadIdx.x;
  float tot = 0.f;
  for (int s = 0; s < 2; ++s) {
    const float* pi = s ? pi1 : pi0;
    const float* pc = s ? pc1 : pc0;
    const float* pb = s ? pb1 : pb0;
    const float* vl = s ? vl1 : vl0;
    float dsum = 0.f;
    for (int c = 0; c < NC; ++c) dsum += vl[b * NC + c];
    float li = 0.f, lc = 0.f;
    for (int c = 0; c < NC; ++c) {
      float v = vl[b * NC + c];
      li += v * logf(pi[b * NC1 + c] + 1e-10f);
      lc += v * logf(pc[b * NC1 + c] + 1e-10f);
    }
    lc += logf(pc[b * NC1 + NC] + 1e-10f);
    float lb = logf(pb[b * NC1 + NC] + 1e-10f);
    tot += -(li / dsum) - lc / (dsum + 1.f) - lb;
  }
  float t = bred(tot, red, 64);
  if (b == 0) atomicAdd(ws + W_ACC, t / (2.f * NB));
}

// guide + sparse losses: one block per (suffix, batch)
__global__ __launch_bounds__(256) void k_guide(const float* cas0,
                                               const float* att0,
                                               const float* cas1,
                                               const float* att1, float* ws) {
  __shared__ float red[256];
  int s = blockIdx.x >> 6;
  int b = blockIdx.x & 63;
  const float* cas = s ? cas1 : cas0;
  const float* att = s ? att1 : att0;
  float g = 0.f, sp = 0.f;
  for (int t = threadIdx.x; t < TT; t += 256) {
    long base = (long)b * TT + t;
    float c = cas[base * NC1 + NC];
    float a0 = att[base * 3 + 0];
    float a1 = att[base * 3 + 1];
    g += fabsf(1.f - c - a0);
    sp += a0 + a1;
  }
  float gt = bred(g, red, 256);
  float spt = bred(sp, red, 256);
  if (threadIdx.x == 0)
    atomicAdd(ws + W_ACC, (LAMB1 * gt + LAMB3 * 0.5f * spt) / (2.f * NB));
}

// feature-norm margin loss: one block per (suffix, batch)
__global__ __launch_bounds__(256) void k_feat(const float* ai0, const float* ac0,
                                              const float* ab0, const float* ai1,
                                              const float* ac1, const float* ab1,
                                              float* ws) {
  __shared__ float red[256];
  int s = blockIdx.x >> 6;
  int b = blockIdx.x & 63;
  const float* A = s ? ai1 : ai0;
  const float* C = s ? ac1 : ac0;
  const float* K = s ? ab1 : ab0;
  float sa = 0.f, sc = 0.f, sb = 0.f;
  for (int i = threadIdx.x; i < DD; i += 256) {
    float x = A[(long)b * DD + i]; sa += x * x;
    float y = C[(long)b * DD + i]; sc += y * y;
    float z = K[(long)b * DD + i]; sb += z * z;
  }
  float ta = bred(sa, red, 256);
  float tc = bred(sc, red, 256);
  float tb = bred(sb, red, 256);
  if (threadIdx.x == 0) {
    float ni = sqrtf(ta), nc = sqrtf(tc), nb = sqrtf(tb);
    float f1 = fmaxf(50.f - ni + nc, 0.f);
    float f2 = fmaxf(50.f - nc + nb, 0.f);
    float f = f1 + f2 + nb;
    atomicAdd(ws + W_ACC, LAMB2 * f * f / (2.f * NB));
  }
}

// LCS: one block per batch. 8 waves build 64x64 cos-sim via f32 WMMA,
// then a 64-wide anti-diagonal wavefront DP in LDS.
__global__ __launch_bounds__(256) void k_lcs(const float* __restrict__ lcs0,
                                             const float* __restrict__ lcs1,
                                             float* ws) {
  __shared__ float S[64][65];
  __shared__ float dp[65][66];
  int b = blockIdx.x;
  int wave = threadIdx.x >> 5, lane = threadIdx.x & 31;
  const float* n0 = ws + W_LCSN0 + b * 64;
  const float* n1 = ws + W_LCSN1 + b * 64;

  for (int tt = wave; tt < 16; tt += 8) {
    int mt = tt >> 2, nt = tt & 3;
    int koff = (lane >> 4) * 2;
    const float* r0 = lcs0 + ((long)b * 64 + mt * 16 + (lane & 15)) * DD + koff;
    const float* r1 = lcs1 + ((long)b * 64 + nt * 16 + (lane & 15)) * DD + koff;
    v8f acc = {};
    for (int k = 0; k < DD; k += 4) {
      v2f a = *(const v2f*)(r0 + k);
      v2f bb = *(const v2f*)(r1 + k);
      acc = wmma4(a, bb, acc);
    }
    int gn = nt * 16 + (lane & 15);
    float nn1 = n1[gn];
    int mbase = mt * 16 + ((lane < 16) ? 0 : 8);
#pragma unroll
    for (int r = 0; r < 8; ++r) {
      int gm = mbase + r;
      S[gm][gn] = acc[r] / (n0[gm] * nn1);
    }
  }
  __syncthreads();
  for (int i = threadIdx.x; i < 65 * 66; i += 256) ((float*)dp)[i] = 0.f;
  __syncthreads();
  for (int d = 2; d <= 128; ++d) {
    if (threadIdx.x < 64) {
      int j = threadIdx.x + 1;
      int i = d - j;
      if (i >= 1 && i <= 64) {
        float s = S[i - 1][j - 1];
        float v = (s > THRES) ? (dp[i - 1][j - 1] + s)
                              : fmaxf(dp[i - 1][j], dp[i][j - 1]);
        dp[i][j] = v;
      }
    }
    __syncthreads();
  }
  if (threadIdx.x == 0) ws[W_LCSV + b] = dp[64][64];
}

// FSD: one block per batch. 4 waves build m,g (32x32) via f32 WMMA over the
// two 1024-wide halves, then a 32-wide wavefront soft-DP (logsumexp).
__global__ __launch_bounds__(128) void k_fsd(const float* __restrict__ c0,
                                             const float* __restrict__ a1,
                                             const float* __restrict__ bk1,
                                             float* ws) {
  __shared__ float M[32][33];
  __shared__ float G[32][33];
  __shared__ float dp[33][34];
  int b = blockIdx.x;
  float pl = ws[W_PL + b];
  const float* c1 = (pl > 0.f) ? a1 : bk1;
  const float* nrm0 = ws + W_FSDN + 0 * 4096 + b * 64;  // (row*2+half)
  const float* nrm1 = ws + W_FSDN + ((pl > 0.f) ? 1 : 2) * 4096 + b * 64;
  int wave = threadIdx.x >> 5, lane = threadIdx.x & 31;

  for (int tt = wave; tt < 8; tt += 4) {
    int half = tt >> 2;
    int mt = (tt >> 1) & 1, nt = tt & 1;
    int koff = half * 1024 + (lane >> 4) * 2;
    const float* r0 = c0 + ((long)b * 32 + mt * 16 + (lane & 15)) * DD + koff;
    const float* r1 = c1 + ((long)b * 32 + nt * 16 + (lane & 15)) * DD + koff;
    v8f acc = {};
    for (int k = 0; k < 1024; k += 4) {
      v2f a = *(const v2f*)(r0 + k);
      v2f bb = *(const v2f*)(r1 + k);
      acc = wmma4(a, bb, acc);
    }
    int gn = nt * 16 + (lane & 15);
    float n1v = nrm1[gn * 2 + half];
    int mbase = mt * 16 + ((lane < 16) ? 0 : 8);
    float(*Dst)[33] = half ? G : M;
#pragma unroll
    for (int r = 0; r < 8; ++r) {
      int gm = mbase + r;
      Dst[gm][gn] = acc[r] / (nrm0[gm * 2 + half] * n1v);
    }
  }
  __syncthreads();
  for (int i = threadIdx.x; i < 33 * 34; i += 128) ((float*)dp)[i] = 0.f;
  __syncthreads();
  for (int d = 2; d <= 64; ++d) {
    if (threadIdx.x < 32) {
      int j = threadIdx.x + 1;
      int i = d - j;
      if (i >= 1 && i <= 32) {
        float a0 = dp[i - 1][j - 1];
        float gij = G[i - 1][j - 1];
        float a1v = gij + dp[i - 1][j];
        float a2v = gij + dp[i][j - 1];
        float hi = fmaxf(a0, fmaxf(a1v, a2v));
        float e = expf((a0 - hi) / GAMMA) + expf((a1v - hi) / GAMMA) +
                  expf((a2v - hi) / GAMMA);
        dp[i][j] = M[i - 1][j - 1] + hi + GAMMA * logf(e);
      }
    }
    __syncthreads();
  }
  if (threadIdx.x == 0) ws[W_FSDV + b] = dp[32][32];
}

__global__ void k_final(const float* ws, float* out) {
  __shared__ float red[64];
  int b = threadIdx.x;
  float pl = ws[W_PL + b];
  float lv = ws[W_LCSV + b];
  float fv = ws[W_FSDV + b];
  float pn = bred(pl, red, 64);
  float posl = bred(lv * pl, red, 64);
  float negl = bred(lv * (1.f - pl), red, 64);
  float posf = bred(fv * pl, red, 64);
  float negf = bred(fv * (1.f - pl), red, 64);
  if (b == 0) {
    float nn = (float)NB - pn;
    float lcs_loss = negl / (nn + 1e-10f) - posl / (pn + 1e-10f);
    float fsd_loss = negf / (nn + 1e-10f) - posf / (pn + 1e-10f);
    out[0] = ws[W_ACC] + LAMB_LCS * lcs_loss + LAMB_FSD * fsd_loss;
  }
}

// ---------------- launch ----------------
extern "C" void kernel_launch(void* const* d_in, const int* in_sizes, int n_in,
                              void* d_out, int out_size, void* d_ws,
                              size_t ws_size, hipStream_t stream) {
  // suffix 0
  const float* pi0 = (const float*)d_in[0];
  const float* pc0 = (const float*)d_in[1];
  const float* pb0 = (const float*)d_in[2];
  const float* vl0 = (const float*)d_in[3];
  const float* att0 = (const float*)d_in[4];
  const float* fi0 = (const float*)d_in[5];
  const float* fc0 = (const float*)d_in[6];
  const float* fb0 = (const float*)d_in[7];
  const float* cas0 = (const float*)d_in[8];
  const float* lcs0 = (const float*)d_in[9];
  const float* fsda0 = (const float*)d_in[10];
  // fsd_bak_candi_0 (d_in[11]) is unused by the reference
  // suffix 1
  const float* pi1 = (const float*)d_in[12];
  const float* pc1 = (const float*)d_in[13];
  const float* pb1 = (const float*)d_in[14];
  const float* vl1 = (const float*)d_in[15];
  const float* att1 = (const float*)d_in[16];
  const float* fi1 = (const float*)d_in[17];
  const float* fc1 = (const float*)d_in[18];
  const float* fb1 = (const float*)d_in[19];
  const float* cas1 = (const float*)d_in[20];
  const float* lcs1 = (const float*)d_in[21];
  const float* fsda1 = (const float*)d_in[22];
  const float* fsdb1 = (const float*)d_in[23];

  float* ws = (float*)d_ws;
  float* out = (float*)d_out;

  k_init<<<1, 64, 0, stream>>>(ws);
  k_pair<<<1, 64, 0, stream>>>(vl0, vl1, ws);
  k_norms<<<2560, 256, 0, stream>>>(lcs0, lcs1, fsda0, fsda1, fsdb1, ws);
  k_cls<<<1, 64, 0, stream>>>(pi0, pc0, pb0, vl0, pi1, pc1, pb1, vl1, ws);
  k_guide<<<128, 256, 0, stream>>>(cas0, att0, cas1, att1, ws);
  k_feat<<<128, 256, 0, stream>>>(fi0, fc0, fb0, fi1, fc1, fb1, ws);
  k_lcs<<<64, 256, 0, stream>>>(lcs0, lcs1, ws);
  k_fsd<<<64, 128, 0, stream>>>(fsda0, fsda1, fsdb1, ws);
  k_final<<<1, 64, 0, stream>>>(ws, out);
}